// EwaldProjector_59236188947234
// MI455X (gfx1250) — compile-verified
//
#include <hip/hip_runtime.h>
#include <math.h>

// ---------------------------------------------------------------------------
// EwaldProjector for MI455X (gfx1250, wave32)
//
//   out[b] = fftshift(ifft2(ifftshift(samp[b]))).real
//
// Shifts folded into sign-modulated DFT matrices:
//   Cs[n,m] = cos(2*pi*n*m/256) * (-1)^(n+m) / 256
//   Ss[n,m] = sin(2*pi*n*m/256) * (-1)^(n+m) / 256
//   Tr = Cs @ X ; Ti = Ss @ X ; out = Tr @ Cs - Ti @ Ss
// GEMMs on V_WMMA_F32_16X16X4_F32 with 2x2 register tiling (32x32 per wave)
// to double arithmetic intensity out of L2.
// ---------------------------------------------------------------------------

typedef float v2f __attribute__((ext_vector_type(2)));
typedef float v8f __attribute__((ext_vector_type(8)));

#define SLEN   256
#define NBATCH 32
#define EWRAD  8.0f

// ---------------------------------------------------------------------------
// Kernel 0: build the sign-folded IDFT matrices (Ss stored directly after Cs
// so that [Cs;Ss] is one contiguous 512x256 matrix for pass 1).
// ---------------------------------------------------------------------------
__global__ void ewald_build_dft(float* __restrict__ Cs, float* __restrict__ Ss) {
  int t = blockIdx.x * blockDim.x + threadIdx.x;   // 65536 threads
  int n = t >> 8;
  int m = t & 255;
  int ph = (n * m) & 255;                          // period-256 of the twiddle
  float ang = (float)ph * (6.283185307179586f / 256.0f);
  float sc  = (((n + m) & 1) ? -1.0f : 1.0f) * (1.0f / 256.0f);
  Cs[t] = sc * cosf(ang);
  Ss[t] = sc * sinf(ang);
}

// ---------------------------------------------------------------------------
// Kernel 1: rotate Ewald-sphere coords and trilinear-sample the volume.
// vol (64MB) is fully L2-resident on MI455X (192MB L2) -> gather-latency bound.
// ---------------------------------------------------------------------------
__global__ void ewald_sample(const float* __restrict__ rot,
                             const float* __restrict__ vol,
                             float* __restrict__ X) {
  int t = blockIdx.x * blockDim.x + threadIdx.x;   // NBATCH*65536 threads
  int b = t >> 16;
  int n = t & 65535;
  int i = n >> 8;
  int j = n & 255;

  const float step = 2.0f / 255.0f;
  float xv = fmaf((float)i, step, -1.0f);          // lin[i] (meshgrid 'ij' x)
  float yv = fmaf((float)j, step, -1.0f);          // lin[j] (meshgrid 'ij' y)
  float r2 = xv * xv + yv * yv;
  float z  = EWRAD - sqrtf(EWRAD * EWRAD - r2);

  // coords = (yv, xv, z); row-vector convention: g_j = sum_i coords_i * R[i][j]
  const float* Rm = rot + b * 9;
  float gx = yv * Rm[0] + xv * Rm[3] + z * Rm[6];
  float gy = yv * Rm[1] + xv * Rm[4] + z * Rm[7];
  float gz = yv * Rm[2] + xv * Rm[5] + z * Rm[8];

  // align_corners=True mapping; x indexes W, y indexes H, z indexes D
  float fx = (gx + 1.0f) * 0.5f * 255.0f;
  float fy = (gy + 1.0f) * 0.5f * 255.0f;
  float fz = (gz + 1.0f) * 0.5f * 255.0f;
  float x0f = floorf(fx), y0f = floorf(fy), z0f = floorf(fz);
  int   x0 = (int)x0f,    y0 = (int)y0f,    z0 = (int)z0f;
  float tx = fx - x0f,    ty = fy - y0f,    tz = fz - z0f;

  float acc = 0.0f;
#pragma unroll
  for (int dz = 0; dz < 2; ++dz) {
    int   zi = z0 + dz;
    float wz = dz ? tz : 1.0f - tz;
    if (zi < 0 || zi > 255) continue;
#pragma unroll
    for (int dy = 0; dy < 2; ++dy) {
      int   yi = y0 + dy;
      float wy = dy ? ty : 1.0f - ty;
      if (yi < 0 || yi > 255) continue;
#pragma unroll
      for (int dx = 0; dx < 2; ++dx) {
        int   xi = x0 + dx;
        float wx = dx ? tx : 1.0f - tx;
        if (xi < 0 || xi > 255) continue;
        acc = fmaf(wx * wy * wz, vol[(zi << 16) + (yi << 8) + xi], acc);
      }
    }
  }
  X[t] = acc;  // X[b][i][j] == projection[b,0,i,j]
}

// ---------------------------------------------------------------------------
// Wave-level 32x32 output block = 2x2 WMMA tiles on V_WMMA_F32_16X16X4_F32.
// Per K-step: 2 A fragments + 2 B fragments feed 4 WMMAs (each fragment
// reused twice -> half the L2 traffic of 1-tile-per-wave).
// ISA f32 fragment layout:
//   A 16x4 : lanes 0-15 -> {K=0,K=1}, lanes 16-31 -> {K=2,K=3}, row = lane%16
//   B 4x16 : mirrored (row k striped across lanes within a VGPR)
//   C/D    : vgpr r -> row r (lanes 0-15) / row r+8 (lanes 16-31), col = lane%16
// acc[rt*2+ct] accumulates tile (rowBase+16*rt, colBase+16*ct).
// ---------------------------------------------------------------------------
template <bool NEG_B>
__device__ __forceinline__ void wmma_block2x2(const float* __restrict__ A, int lda,
                                              const float* __restrict__ Bm, int ldb,
                                              int K, int rowBase, int colBase,
                                              v8f acc[4]) {
  const int lane = threadIdx.x & 31;
  const int half = lane >> 4;        // 0: K pair {0,1}, 1: K pair {2,3}
  const int lid  = lane & 15;
  const float* arow0 = A + (rowBase + lid) * lda + 2 * half;
  const float* arow1 = arow0 + 16 * lda;
  const float* bcol0 = Bm + colBase + lid + 2 * half * ldb;
  const float* bcol1 = bcol0 + 16;

  for (int k = 0; k < K; k += 4) {
    v2f a0, a1, b0, b1;
    a0.x = arow0[k];
    a0.y = arow0[k + 1];
    a1.x = arow1[k];
    a1.y = arow1[k + 1];
    const float* bp0 = bcol0 + k * ldb;
    const float* bp1 = bcol1 + k * ldb;
    b0.x = bp0[0];
    b0.y = bp0[ldb];
    b1.x = bp1[0];
    b1.y = bp1[ldb];
    if (NEG_B) {
      b0.x = -b0.x; b0.y = -b0.y;
      b1.x = -b1.x; b1.y = -b1.y;
    }
    // (neg_a, A, neg_b, B, c_mod, C, reuse_a, reuse_b)
    acc[0] = __builtin_amdgcn_wmma_f32_16x16x4_f32(false, a0, false, b0,
                                                   (short)0, acc[0], false, false);
    acc[1] = __builtin_amdgcn_wmma_f32_16x16x4_f32(false, a0, false, b1,
                                                   (short)0, acc[1], false, false);
    acc[2] = __builtin_amdgcn_wmma_f32_16x16x4_f32(false, a1, false, b0,
                                                   (short)0, acc[2], false, false);
    acc[3] = __builtin_amdgcn_wmma_f32_16x16x4_f32(false, a1, false, b1,
                                                   (short)0, acc[3], false, false);
  }
}

__device__ __forceinline__ void store_block2x2(float* __restrict__ D, int ldd,
                                               int rowBase, int colBase,
                                               const v8f acc[4]) {
  const int lane = threadIdx.x & 31;
  const int half = lane >> 4;
  const int lid  = lane & 15;
#pragma unroll
  for (int rt = 0; rt < 2; ++rt) {
#pragma unroll
    for (int ct = 0; ct < 2; ++ct) {
      float* dp = D + (rowBase + 16 * rt + half * 8) * ldd + colBase + 16 * ct + lid;
      const v8f& c = acc[rt * 2 + ct];
#pragma unroll
      for (int r = 0; r < 8; ++r) dp[r * ldd] = c[r];
    }
  }
}

// Pass 1: TT[b] (512x256) = [Cs;Ss] (512x256) @ X[b] (256x256)
__global__ void ewald_gemm1(const float* __restrict__ Wstk,  // [Cs;Ss]
                            const float* __restrict__ X,
                            float* __restrict__ TT) {
  const int rowBase = blockIdx.x * 32;   // 16 row blocks of 512
  const int colBase = blockIdx.y * 32;   // 8 col blocks of 256
  const int b       = blockIdx.z;
  v8f acc[4] = {};
  wmma_block2x2<false>(Wstk, 256, X + b * 65536, 256, 256, rowBase, colBase, acc);
  store_block2x2(TT + b * 512 * 256, 256, rowBase, colBase, acc);
}

// Pass 2: out[b] = Tr[b] @ Cs - Ti[b] @ Ss
__global__ void ewald_gemm2(const float* __restrict__ Cs,
                            const float* __restrict__ Ss,
                            const float* __restrict__ TT,
                            float* __restrict__ Out) {
  const int rowBase = blockIdx.x * 32;   // 8 row blocks of 256
  const int colBase = blockIdx.y * 32;   // 8 col blocks of 256
  const int b       = blockIdx.z;
  const float* Tr = TT + b * 512 * 256;
  const float* Ti = Tr + 256 * 256;
  v8f acc[4] = {};
  wmma_block2x2<false>(Tr, 256, Cs, 256, 256, rowBase, colBase, acc);
  wmma_block2x2<true >(Ti, 256, Ss, 256, 256, rowBase, colBase, acc);
  store_block2x2(Out + b * 65536, 256, rowBase, colBase, acc);
}

// ---------------------------------------------------------------------------
extern "C" void kernel_launch(void* const* d_in, const int* in_sizes, int n_in,
                              void* d_out, int out_size, void* d_ws, size_t ws_size,
                              hipStream_t stream) {
  (void)in_sizes; (void)n_in; (void)out_size; (void)ws_size;

  const float* rotmat = (const float*)d_in[0];  // [32,3,3] f32
  const float* vol    = (const float*)d_in[1];  // [256^3]  f32
  float*       out    = (float*)d_out;          // [32,1,256,256] f32

  // Workspace layout (floats): Cs | Ss | X | TT  (~25.7 MB total)
  float* Cs = (float*)d_ws;                       // 65536
  float* Ss = Cs + 65536;                         // 65536 (contiguous -> [Cs;Ss])
  float* X  = Ss + 65536;                         // 32*65536
  float* TT = X + NBATCH * 65536;                 // 32*512*256

  ewald_build_dft<<<dim3(256), dim3(256), 0, stream>>>(Cs, Ss);
  ewald_sample<<<dim3((NBATCH * 65536) / 256), dim3(256), 0, stream>>>(rotmat, vol, X);

  dim3 g1(16, 8, NBATCH);
  ewald_gemm1<<<g1, dim3(32), 0, stream>>>(Cs /* [Cs;Ss] stacked */, X, TT);

  dim3 g2(8, 8, NBATCH);
  ewald_gemm2<<<g2, dim3(32), 0, stream>>>(Cs, Ss, TT, out);
}